// I7Pool_326417514939
// MI455X (gfx1250) — compile-verified
//
#include <hip/hip_runtime.h>

// Directional (h'>=h, w'>=w) guided max pooling on MI455X (gfx1250):
//   v = x * guide (guide broadcast over C=256), out = revcummax_H(revcummax_W(v))
// One workgroup per (b,c) 128x128 fp32 plane; 4096 workgroups; 128 threads
// (4 wave32s). Bandwidth-bound: ~538 MB total -> ~23 us floor @ 23.3 TB/s.
//
// CDNA5 path: the whole x-plane is brought into LDS by ONE Tensor Data Mover
// DMA (TENSOR_LOAD_TO_LDS, TENSORcnt) with pad_enable inserting 1 DWORD of
// padding every 128 DWORDs -> stride-129 LDS rows, which are bank-conflict-
// free for BOTH the H-direction scan (consecutive banks across lanes) and the
// W-direction scan (bank = (t+w)%64, all distinct). Guide is warmed with
// global_prefetch while the DMA streams.

#define PLANE_H   128
#define PLANE_W   128
#define NCH       256
#define LDS_STRIDE 129                       // 128 data + 1 pad DWORD per row
#define LDS_BYTES (PLANE_H * LDS_STRIDE * 4) // 66048 B (dynamic LDS, <320KB/WGP)

typedef unsigned int u32x4 __attribute__((ext_vector_type(4)));
typedef int          i32x8 __attribute__((ext_vector_type(8)));
typedef int          i32x4 __attribute__((ext_vector_type(4)));

__global__ __launch_bounds__(128)
void I7Pool_dirmax_kernel(const float* __restrict__ x,
                          const float* __restrict__ g,
                          float* __restrict__ out) {
    extern __shared__ float tile[];          // PLANE_H * LDS_STRIDE floats

    const unsigned t     = threadIdx.x;      // 0..127
    const unsigned plane = blockIdx.x;       // b*256 + c
    const unsigned b     = plane >> 8;       // / NCH

    const float* xp = x   + (size_t)plane * (PLANE_H * PLANE_W);
    const float* gp = g   + (size_t)b     * (PLANE_H * PLANE_W);
    float*       op = out + (size_t)plane * (PLANE_H * PLANE_W);

    // Wave-relative LDS byte offset (flat-LDS addressing: low 32 bits of the
    // generic pointer are the offset the TDM / async units consume).
    const unsigned lds_base = (unsigned)(uintptr_t)(const void*)&tile[0];

    // Warm L2 with the guide plane (one prefetch per 128B line) while the
    // tensor DMA streams the x plane. 128 thr * 32 floats * 4 iter = plane.
    #pragma unroll
    for (unsigned j = 0; j < 4u; ++j)
        __builtin_prefetch(&gp[j * 4096u + t * 32u], 0, 3);

#if __has_builtin(__builtin_amdgcn_tensor_load_to_lds)
    // ---------- Pass 0: one TDM DMA for the whole 64KB plane --------------
    if (t < 32u) {                           // wave 0 issues the single DMA
        const unsigned long long ga = (unsigned long long)(uintptr_t)xp;
        // D# group 0: count=1 | lds_addr | global_addr[56:0] | type=2
        u32x4 g0;
        g0.x = 1u;                                        // count=1, user D#
        g0.y = lds_base;                                  // lds_addr (bytes)
        g0.z = (unsigned)ga;                              // global_addr[31:0]
        g0.w = (unsigned)(ga >> 32) | (2u << 30);         // [56:32] | type=2
        // D# group 1: data_size=4B, pad 1 DWORD every 128 DWORDs,
        // tensor 128x128, tile 128x128, dim0_stride=128 (data_size units)
        i32x8 g1;
        g1[0] = (int)((2u << 16) | (1u << 20) | (6u << 22)); // dsz|pad_en|intv
        g1[1] = (int)(128u << 16);   // tensor_dim0[15:0]=128 (atomic_addr=0)
        g1[2] = (int)(128u << 16);   // tensor_dim0 hi=0 | tensor_dim1=128
        g1[3] = (int)(128u << 16);   // tensor_dim1 hi=0 | tile_dim0=128
        g1[4] = 128;                 // tile_dim1=128, tile_dim2=0(unused)
        g1[5] = 128;                 // tensor_dim0_stride[31:0]=128
        g1[6] = (int)(0x4000u << 16);// dim0_stride hi=0 | dim1_stride lo=16384
        g1[7] = 0;                   // dim1_stride hi (ignored: 2-D tile)
        const i32x4 gz4 = {0, 0, 0, 0};          // groups 2/3: unused (2-D)
        const i32x8 gz8 = {0, 0, 0, 0, 0, 0, 0, 0};
        // clang-23 lane: 6-arg form (u32x4, i32x8, i32x4, i32x4, i32x8, cpol)
        __builtin_amdgcn_tensor_load_to_lds(g0, g1, gz4, gz4, gz8, 0);
        __builtin_amdgcn_s_wait_tensorcnt(0);
    }
#else
    // ---------- Fallback: async global->LDS per-lane scatter --------------
    #pragma unroll 8
    for (unsigned h = 0; h < PLANE_H; ++h) {
        const unsigned goff = (h * 128u + t) * 4u;
        const unsigned loff = lds_base + (h * LDS_STRIDE + t) * 4u;
        asm volatile("global_load_async_to_lds_b32 %0, %1, %2"
                     :: "v"(loff), "v"(goff), "s"(xp) : "memory");
    }
    asm volatile("s_wait_asynccnt 0x0" ::: "memory");
#endif
    __syncthreads();                         // DMA-complete + cross-wave vis

    // ---------- Pass 1: reverse cummax along H (thread t = column t) ------
    {
        float carry = -__builtin_inff();
        #pragma unroll 4
        for (int h = PLANE_H - 1; h >= 0; --h) {
            const unsigned i  = (unsigned)h * LDS_STRIDE + t;   // banks t+.. ok
            const float    gv = gp[(unsigned)h * 128u + t];     // coalesced L2
            carry = fmaxf(carry, tile[i] * gv);
            tile[i] = carry;
        }
    }
    __syncthreads();

    // ---------- Pass 2: reverse cummax along W (thread t = row t) ---------
    {
        float carry = -__builtin_inff();
        #pragma unroll 4
        for (int w = PLANE_W - 1; w >= 0; --w) {
            const unsigned i = t * LDS_STRIDE + (unsigned)w;    // (t+w)%64 ok
            carry = fmaxf(carry, tile[i]);
            tile[i] = carry;
        }
    }
    __syncthreads();

    // ---------- Pass 3: coalesced float4 store ----------------------------
    {
        const unsigned w = (t & 31u) * 4u;       // float4 group within row
        const unsigned r = t >> 5u;              // wave id 0..3
        #pragma unroll 2
        for (unsigned j = 0; j < 32u; ++j) {
            const unsigned h = j * 4u + r;       // each wave stores full rows
            const unsigned i = h * LDS_STRIDE + w;
            float4 v;                            // LDS row not 16B-aligned ->
            v.x = tile[i + 0]; v.y = tile[i + 1];//  4 scalar ds reads
            v.z = tile[i + 2]; v.w = tile[i + 3];
            *(float4*)&op[h * 128u + w] = v;     // coalesced 512B/wave store
        }
    }
}

extern "C" void kernel_launch(void* const* d_in, const int* in_sizes, int n_in,
                              void* d_out, int out_size, void* d_ws, size_t ws_size,
                              hipStream_t stream) {
    (void)in_sizes; (void)n_in; (void)d_ws; (void)ws_size; (void)out_size;
    const float* x   = (const float*)d_in[0];   // (16,256,128,128) f32
    const float* gde = (const float*)d_in[1];   // (16,  1,128,128) f32
    float*       out = (float*)d_out;           // (16,256,128,128) f32

    const int planes = 16 * NCH;                // 4096 workgroups
    I7Pool_dirmax_kernel<<<dim3(planes), dim3(128), LDS_BYTES, stream>>>(x, gde, out);
}